// EPHNLoss_1915555414690
// MI455X (gfx1250) — compile-verified
//
#include <hip/hip_runtime.h>
#include <hip/hip_bf16.h>

// EPHN loss, fused GEMM (fn @ fn^T) + easy-positive / hard-negative mining.
// f32 WMMA (v_wmma_f32_16x16x4_f32) keeps reference precision; data is L2-resident.

typedef __attribute__((ext_vector_type(2))) float v2f;
typedef __attribute__((ext_vector_type(8))) float v8f;

#define DVEC 128          // feature dim (reference: D=128)
#define LDSTR 132         // padded LDS row stride (floats) -> conflict-free ds_load_b64
#define INV_SIGMA 10.0f   // 1 / 0.1

// ---------------------------------------------------------------------------
// Kernel 1: per-row inverse L2 norm. One wave per row, float4 loads.
// ---------------------------------------------------------------------------
__global__ void ephn_norm_kernel(const float* __restrict__ fvec,
                                 float* __restrict__ rnorm) {
    const int row  = blockIdx.x * 4 + (threadIdx.x >> 5);
    const int lane = threadIdx.x & 31;
    const float4 d = ((const float4*)(fvec + ((size_t)row << 7)))[lane];
    float s = d.x * d.x + d.y * d.y + d.z * d.z + d.w * d.w;
    #pragma unroll
    for (int off = 16; off >= 1; off >>= 1)
        s += __shfl_xor(s, off, 32);
    if (lane == 0) rnorm[row] = rsqrtf(s);
}

// ---------------------------------------------------------------------------
// Kernel 2: fused similarity GEMM + masked row max/argmax mining.
//   block = 128 threads (4 waves), owns 16 rows (i-block).
//   Sweeps 64-column strips; each wave computes one 16x16 tile via
//   32 chained v_wmma_f32_16x16x4_f32 (K = 128).
// ---------------------------------------------------------------------------
__global__ void ephn_main_kernel(const float* __restrict__ fvec,
                                 const int*   __restrict__ Lvec,
                                 const float* __restrict__ rnorm,
                                 float* __restrict__ wsPos,
                                 float* __restrict__ wsNeg,
                                 int N) {
    __shared__ float lds_a[16 * LDSTR];
    __shared__ float lds_b[64 * LDSTR];
    __shared__ float mPV[64]; __shared__ int mPJ[64];
    __shared__ float mNV[64]; __shared__ int mNJ[64];

    const int tid = threadIdx.x;
    const int l   = tid & 31;    // lane in wave
    const int w   = tid >> 5;    // wave id 0..3
    const int m   = l & 15;      // row (A) / column (B) index within tile
    const int hi  = l >> 4;      // half-wave select
    const int i0  = blockIdx.x * 16;

    // ---- stage normalized A panel (16 rows x 128) into LDS, padded rows ----
    for (int t = tid; t < 16 * 32; t += 128) {      // 512 float4 chunks
        const int f   = t * 4;
        const int row = f >> 7;
        const int col = f & 127;
        float4 d = *(const float4*)(fvec + ((size_t)i0 << 7) + f);
        const float r = rnorm[i0 + row];
        float* dst = &lds_a[row * LDSTR + col];
        dst[0] = d.x * r; dst[1] = d.y * r; dst[2] = d.z * r; dst[3] = d.w * r;
    }

    // row labels + row ids for this lane's 8 accumulator rows
    int Li[8];
    #pragma unroll
    for (int v = 0; v < 8; ++v) Li[v] = Lvec[i0 + v + 8 * hi];

    // per-(row,lane-column) running bests; tie-break: smallest j (argmax-first)
    float posV[8], negV[8]; int posJ[8], negJ[8];
    #pragma unroll
    for (int v = 0; v < 8; ++v) {
        posV[v] = -2.0f; negV[v] = -2.0f;
        posJ[v] = 0x7fffffff; negJ[v] = 0x7fffffff;
    }

    const float* aptr = &lds_a[m * LDSTR + 2 * hi];
    const float* bptr = &lds_b[(w * 16 + m) * LDSTR + 2 * hi];

    const int nstrips = N >> 6;   // 64 columns per strip
    for (int jt = 0; jt < nstrips; ++jt) {
        __syncthreads();  // previous strip fully consumed before overwrite
        // ---- stage normalized B panel (64 rows x 128) into LDS ----
        const size_t gbase = ((size_t)jt << 6) << 7;   // jt*64*128
        for (int t = tid; t < 64 * 32; t += 128) {     // 2048 float4 chunks
            const int f   = t * 4;
            const int row = f >> 7;
            const int col = f & 127;
            float4 d = *(const float4*)(fvec + gbase + f);
            const float r = rnorm[jt * 64 + row];
            float* dst = &lds_b[row * LDSTR + col];
            dst[0] = d.x * r; dst[1] = d.y * r; dst[2] = d.z * r; dst[3] = d.w * r;
        }
        __syncthreads();

        // ---- 16x16 tile: C = A(16xK) * B(Kx16), K=128, 32 wmma steps ----
        v8f c = {};
        #pragma unroll
        for (int k = 0; k < DVEC; k += 4) {
            v2f a = *(const v2f*)(aptr + k);   // ds_load_b64 (conflict-free)
            v2f b = *(const v2f*)(bptr + k);
            c = __builtin_amdgcn_wmma_f32_16x16x4_f32(
                    false, a, false, b, (short)0, c, false, false);
        }

        // ---- masks + running max/argmax update ----
        const int j  = jt * 64 + w * 16 + m;   // global column for this lane
        const int Lj = Lvec[j];
        #pragma unroll
        for (int v = 0; v < 8; ++v) {
            const int   i   = i0 + v + 8 * hi;
            const float d   = c[v];
            const bool  eqc = (Li[v] == Lj);
            // easy-positive candidate: same class, not diag, not >0.9999
            const float pv = eqc ? (((i == j) || (d > 0.9999f)) ? -1.0f : d)
                                 : -1.0f;
            if (pv > posV[v] || (pv == posV[v] && j < posJ[v])) {
                posV[v] = pv; posJ[v] = j;
            }
            // hard-negative candidate: different class (diag -> -1 via eqc)
            const float nv = eqc ? -1.0f : d;
            if (nv > negV[v] || (nv == negV[v] && j < negJ[v])) {
                negV[v] = nv; negJ[v] = j;
            }
        }
    }

    // ---- reduce across the 16 column-lanes of each half-wave ----
    #pragma unroll
    for (int v = 0; v < 8; ++v) {
        #pragma unroll
        for (int off = 1; off < 16; off <<= 1) {
            float ov = __shfl_xor(posV[v], off, 16);
            int   oj = __shfl_xor(posJ[v], off, 16);
            if (ov > posV[v] || (ov == posV[v] && oj < posJ[v])) {
                posV[v] = ov; posJ[v] = oj;
            }
            float onv = __shfl_xor(negV[v], off, 16);
            int   onj = __shfl_xor(negJ[v], off, 16);
            if (onv > negV[v] || (onv == negV[v] && onj < negJ[v])) {
                negV[v] = onv; negJ[v] = onj;
            }
        }
    }

    // ---- merge the 4 waves' per-row results through LDS ----
    if ((l & 15) == 0) {
        #pragma unroll
        for (int v = 0; v < 8; ++v) {
            const int row = v + 8 * hi;
            const int idx = w * 16 + row;
            mPV[idx] = posV[v]; mPJ[idx] = posJ[v];
            mNV[idx] = negV[v]; mNJ[idx] = negJ[v];
        }
    }
    __syncthreads();
    if (tid < 16) {
        float bpv = -2.0f, bnv = -2.0f;
        int   bpj = 0x7fffffff, bnj = 0x7fffffff;
        #pragma unroll
        for (int ww = 0; ww < 4; ++ww) {
            const int idx = ww * 16 + tid;
            const float pv = mPV[idx]; const int pj = mPJ[idx];
            if (pv > bpv || (pv == bpv && pj < bpj)) { bpv = pv; bpj = pj; }
            const float nv = mNV[idx]; const int nj = mNJ[idx];
            if (nv > bnv || (nv == bnv && nj < bnj)) { bnv = nv; bnj = nj; }
        }
        wsPos[i0 + tid] = bpv;
        wsNeg[i0 + tid] = bnv;
    }
}

// ---------------------------------------------------------------------------
// Kernel 3: loss + logging outputs. Single 1024-thread block.
// out layout: [loss | eps(N+2) | hns(N+2) | pnd]  == 2N+6 floats.
// ---------------------------------------------------------------------------
__global__ void ephn_finalize_kernel(const float* __restrict__ wsPos,
                                     const float* __restrict__ wsNeg,
                                     float* __restrict__ out, int N) {
    __shared__ float sProb[1024];
    __shared__ float sPos[1024];
    __shared__ float sNeg[1024];
    __shared__ int   sCnt[1024];
    const int tid = threadIdx.x;

    float* eps = out + 1;
    float* hns = out + 1 + (N + 2);

    float accProb = 0.0f, accPos = 0.0f, accNeg = 0.0f;
    int accCnt = 0;
    for (int i = tid; i < N; i += 1024) {
        const float p = wsPos[i];
        const float n = wsNeg[i];
        eps[i] = (p > 0.0f) ? p : 0.0f;
        hns[i] = (n > 0.0f) ? n : 0.0f;
        accPos += p;
        accNeg += n;
        if (p > 0.0f && n > 0.0f) {
            const float s = (n - p) * INV_SIGMA;
            // -log_softmax([p,n]/sigma)[0] == softplus((n-p)/sigma), stable
            const float prob = (s > 0.0f) ? (s + log1pf(expf(-s)))
                                          : log1pf(expf(s));
            accProb += prob;
            accCnt  += 1;
        }
    }
    sProb[tid] = accProb; sPos[tid] = accPos; sNeg[tid] = accNeg; sCnt[tid] = accCnt;
    __syncthreads();
    for (int st = 512; st > 0; st >>= 1) {
        if (tid < st) {
            sProb[tid] += sProb[tid + st];
            sPos[tid]  += sPos[tid + st];
            sNeg[tid]  += sNeg[tid + st];
            sCnt[tid]  += sCnt[tid + st];
        }
        __syncthreads();
    }
    if (tid == 0) {
        const int cnt = (sCnt[0] > 1) ? sCnt[0] : 1;
        out[0] = sProb[0] / (float)cnt;
        eps[N] = 0.0f; eps[N + 1] = 1.0f;
        hns[N] = 0.0f; hns[N + 1] = 1.0f;
        out[1 + 2 * (N + 2)] = (sPos[0] - sNeg[0]) / (float)N;
    }
}

// ---------------------------------------------------------------------------
extern "C" void kernel_launch(void* const* d_in, const int* in_sizes, int n_in,
                              void* d_out, int out_size, void* d_ws, size_t ws_size,
                              hipStream_t stream) {
    const float* fvec = (const float*)d_in[0];
    const int*   Lvec = (const int*)d_in[1];
    const int N = in_sizes[1];          // 8192 (D fixed at 128 per reference)

    float* rnorm = (float*)d_ws;        // N floats
    float* wsPos = rnorm + N;           // N floats
    float* wsNeg = wsPos + N;           // N floats  (total 96 KB << ws_size)

    ephn_norm_kernel<<<N / 4, 128, 0, stream>>>(fvec, rnorm);
    ephn_main_kernel<<<N / 16, 128, 0, stream>>>(fvec, Lvec, rnorm,
                                                 wsPos, wsNeg, N);
    ephn_finalize_kernel<<<1, 1024, 0, stream>>>(wsPos, wsNeg,
                                                 (float*)d_out, N);
}